// transforms_floyd_steinberg_halftoning_24094766531116
// MI455X (gfx1250) — compile-verified
//
#include <hip/hip_runtime.h>
#include <stdint.h>

// MI455X / gfx1250 Floyd-Steinberg halftoning.
// Memory-bound (~201 MB @ 23.3 TB/s => ~8.6 us floor); no matmul structure,
// so the CDNA5 paths used are the Tensor Data Mover (TENSOR_LOAD_TO_LDS +
// s_wait_tensorcnt) for LDS staging -- with the per-lane async-b128 path
// (global_load_async_to_lds_b128 + s_wait_asynccnt) as fallback -- and
// wave32 per-lane 8x8 tiles. Zero-input masking is fused into the store
// phase by re-reading the still-resident LDS input.

typedef __attribute__((ext_vector_type(4))) float f4;

#define IMG_W 1024
#define IMG_H 1024
#define TILES_Y 128  // IMG_H / 8

#if __has_builtin(__builtin_amdgcn_tensor_load_to_lds) && __has_builtin(__builtin_amdgcn_s_wait_tensorcnt)
#define USE_TDM 1
typedef __attribute__((ext_vector_type(4))) unsigned int u32x4;
typedef __attribute__((ext_vector_type(8))) int i32x8;
typedef __attribute__((ext_vector_type(4))) int i32x4;
#endif

__global__ __launch_bounds__(128, 1) void fs_halftone(const float* __restrict__ in,
                                                      float* __restrict__ out)
{
    // 3 channels x 8 image rows x 1024 pixels = 96 KB (WGP has 320 KB LDS)
    __shared__ float smem[3 * 8 * IMG_W];

    const int t  = threadIdx.x;          // tile column index wb: 0..127
    const int b  = blockIdx.x >> 7;      // batch image
    const int hb = blockIdx.x & 127;     // tile row index
    const int y0 = hb * 8;
    const int x0 = t * 8;

    // ---- Stage 3 channels x 8 contiguous image rows into LDS ----
#ifdef USE_TDM
    // Tensor Data Mover path: each of the 4 waves issues 3 one-shot 1D
    // descriptors (one per channel, 2 rows x 1024 floats = 8 KB each).
    // D# layout per CDNA5 ISA ch.8: group0 {count|flags, lds_addr,
    // global_addr lo, global_addr hi | type=2}; group1 {data_size=4B,
    // tensor_dim0=tile_dim0=2048 floats, all else 0}; groups 2/3 zero (1D).
    {
        const int w = t >> 5;  // wave id in block (uniform per wave)
        #pragma unroll
        for (int c = 0; c < 3; ++c) {
            const float* src = in + (((size_t)(b * 3 + c)) * IMG_H + (size_t)(y0 + 2 * w)) * IMG_W;
            const uint64_t gaddr = (uint64_t)(uintptr_t)src;
            const uint32_t laddr = (uint32_t)(uintptr_t)&smem[c * (8 * IMG_W) + w * 2048];

            u32x4 g0 = {
                1u,                                                   // count=1, user mode
                laddr,                                                // lds_addr (bytes)
                (uint32_t)gaddr,                                      // global_addr[31:0]
                (uint32_t)((gaddr >> 32) & 0x01FFFFFFu) | (2u << 30)  // addr[56:32], type=2
            };
            i32x8 g1 = {
                (int)(2u << 16),       // data_size = 4 bytes
                (int)(2048u << 16),    // tensor_dim0[15:0] = 2048
                0,                     // tensor_dim0 hi = 0, tensor_dim1 = 0
                (int)(2048u << 16),    // tile_dim0 = 2048
                0, 0, 0, 0             // tile_dim1/2 = 0 (1D), strides unused
            };
            i32x4 gz4 = {0, 0, 0, 0};
            i32x8 gz8 = {0, 0, 0, 0, 0, 0, 0, 0};
            __builtin_amdgcn_tensor_load_to_lds(g0, g1, gz4, gz4, gz8, 0);
        }
        __builtin_amdgcn_s_wait_tensorcnt(0);
    }
#else
    // Fallback: per-lane async copies; each b128 moves 512 B coalesced per wave (2 KB per block).
    #pragma unroll
    for (int c = 0; c < 3; ++c) {
        const float* src = in + (((size_t)(b * 3 + c)) * IMG_H + (size_t)y0) * IMG_W;
        const uint32_t lbase = (uint32_t)(uintptr_t)&smem[c * (8 * IMG_W)];
        #pragma unroll
        for (int k = 0; k < 16; ++k) {
            const uint32_t foff  = (uint32_t)(k * 512 + t * 4);      // float index
            const uint64_t gaddr = (uint64_t)(uintptr_t)(src + foff);
            const uint32_t laddr = lbase + foff * 4u;
            asm volatile("global_load_async_to_lds_b128 %0, %1, off"
                         :: "v"(laddr), "v"(gaddr)
                         : "memory");
        }
    }
#if __has_builtin(__builtin_amdgcn_s_wait_asynccnt)
    __builtin_amdgcn_s_wait_asynccnt(0);
#else
    asm volatile("s_wait_asynccnt 0x0" ::: "memory");
#endif
#endif  // USE_TDM
    __syncthreads();

    // ---- Per-lane tile: grayscale from LDS ----
    // g[i][j] = gray at (y = y0+j, x = x0+i); matches reference's (W,H)-transposed scan.
    float g[8][8];

    #pragma unroll
    for (int j = 0; j < 8; ++j) {
        float pc[3][8];
        #pragma unroll
        for (int c = 0; c < 3; ++c) {
            const f4* p = (const f4*)&smem[c * (8 * IMG_W) + j * IMG_W + x0];
            f4 lo = p[0];
            f4 hi = p[1];
            pc[c][0] = lo.x; pc[c][1] = lo.y; pc[c][2] = lo.z; pc[c][3] = lo.w;
            pc[c][4] = hi.x; pc[c][5] = hi.y; pc[c][6] = hi.z; pc[c][7] = hi.w;
        }
        #pragma unroll
        for (int i = 0; i < 8; ++i) {
            const float s0 = pc[0][i] * 255.0f;
            const float s1 = pc[1][i] * 255.0f;
            const float s2 = pc[2][i] * 255.0f;
            g[i][j] = 0.114f * s0 + 0.587f * s1 + 0.299f * s2;
        }
    }

    // ---- Serial Floyd-Steinberg inside the 8x8 tile (fully unrolled) ----
    // Scan rows i (x direction), pixels j (y direction), per the reference transpose:
    // 7/16 -> j+1 (same i), next row i+1 gets 3/16 @ j-1, 5/16 @ j, 1/16 @ j+1.
    float nr[8];
    #pragma unroll
    for (int j = 0; j < 8; ++j) nr[j] = 0.0f;

    #pragma unroll
    for (int i = 0; i < 8; ++i) {
        float e[8];
        float er = 0.0f;
        #pragma unroll
        for (int j = 0; j < 8; ++j) {
            const float old = (g[i][j] + nr[j]) + er;        // (row + carry) + err_right
            const float nw  = (old > 127.0f) ? 1.0f : 0.0f;  // already /255 normalized
            const float ee  = old - nw * 255.0f;
            er      = ee * (7.0f / 16.0f);
            g[i][j] = nw;                                    // overwrite with halftone {0,1}
            e[j]    = ee;
        }
        #pragma unroll
        for (int j = 0; j < 8; ++j) {
            float v = (5.0f / 16.0f) * e[j];
            if (j < 7) v += (3.0f / 16.0f) * e[j + 1];
            if (j > 0) v += (1.0f / 16.0f) * e[j - 1];
            nr[j] = v;
        }
    }

    // ---- Broadcast to 3 channels; fuse zero-input mask by re-reading LDS ----
    #pragma unroll
    for (int c = 0; c < 3; ++c) {
        float* dst = out + (((size_t)(b * 3 + c)) * IMG_H + (size_t)y0) * IMG_W + x0;
        #pragma unroll
        for (int j = 0; j < 8; ++j) {
            const f4* p = (const f4*)&smem[c * (8 * IMG_W) + j * IMG_W + x0];
            const f4 vlo = p[0];
            const f4 vhi = p[1];
            f4 lo, hi;
            lo.x = (vlo.x == 0.0f) ? 0.0f : g[0][j];
            lo.y = (vlo.y == 0.0f) ? 0.0f : g[1][j];
            lo.z = (vlo.z == 0.0f) ? 0.0f : g[2][j];
            lo.w = (vlo.w == 0.0f) ? 0.0f : g[3][j];
            hi.x = (vhi.x == 0.0f) ? 0.0f : g[4][j];
            hi.y = (vhi.y == 0.0f) ? 0.0f : g[5][j];
            hi.z = (vhi.z == 0.0f) ? 0.0f : g[6][j];
            hi.w = (vhi.w == 0.0f) ? 0.0f : g[7][j];
            f4* q = (f4*)(dst + (size_t)j * IMG_W);
            __builtin_nontemporal_store(lo, q);
            __builtin_nontemporal_store(hi, q + 1);
        }
    }
}

extern "C" void kernel_launch(void* const* d_in, const int* in_sizes, int n_in,
                              void* d_out, int out_size, void* d_ws, size_t ws_size,
                              hipStream_t stream) {
    const float* in = (const float*)d_in[0];
    float* out = (float*)d_out;
    const int B = in_sizes[0] / (3 * IMG_H * IMG_W);  // batch = 8 for the reference shapes
    dim3 grid((unsigned)(B * TILES_Y));
    dim3 block(128);
    hipLaunchKernelGGL(fs_halftone, grid, block, 0, stream, in, out);
}